// StrokeFieldPseudoRenderer_57964878627319
// MI455X (gfx1250) — compile-verified
//
#include <hip/hip_runtime.h>
#include <math.h>

#define Bn   8
#define Hn   64
#define Wn   64
#define FCn  768
#define RCn  8
#define Pn   8
#define HWn  (Hn*Wn)          // 4096
#define NPIX (Bn*HWn)         // 32768
#define PI_F 3.14159265358979323846f

typedef __attribute__((ext_vector_type(2))) float v2f;
typedef __attribute__((ext_vector_type(8))) float v8f;

__device__ __forceinline__ float geluf(float x){
  return 0.5f*x*(1.0f+erff(x*0.7071067811865476f));
}
__device__ __forceinline__ float sigmf_(float x){ return 1.0f/(1.0f+expf(-x)); }

// ---------------- K0: build prototype bank + 0.05*proto_delta ----------------
__global__ void k_bank(const float* __restrict__ pd, float* __restrict__ kern){
  __shared__ float sd[128];
  int p = blockIdx.x, t = threadIdx.x;
  bool act = t < 81;
  int i = t/9, j = t%9;
  float yy = -1.0f + 0.25f*(float)i;
  float xx = -1.0f + 0.25f*(float)j;
  float ang = 2.0f*PI_F*(float)p/8.0f;
  const float Ls[3] = {0.45f,0.75f,1.05f};
  const float Ws[3] = {0.14f,0.20f,0.28f};
  float L = Ls[p%3], W = Ws[(p/3)%3];
  float ca = cosf(ang), sa = sinf(ang);
  float xr =  xx*ca + yy*sa;
  float yr = -xx*sa + yy*ca;
  float tpr  = powf(1.0f - fminf(fabsf(xr)/L, 1.0f), 1.5f);
  float core = expf(-0.5f*((xr/L)*(xr/L) + (yr/W)*(yr/W)));
  float ridge = act ? tpr*core*fmaxf(1.0f-(yr/W)*(yr/W), 0.0f) : 0.0f;
  sd[t] = ridge; __syncthreads();
  for (int s=64; s>0; s>>=1){ if (t<s) sd[t]+=sd[t+s]; __syncthreads(); }
  float mean = sd[0]/81.0f; __syncthreads();
  float kv = ridge - mean;
  sd[t] = act ? fabsf(kv) : 0.0f; __syncthreads();
  for (int s=64; s>0; s>>=1){ if (t<s) sd[t]+=sd[t+s]; __syncthreads(); }
  float asum = fmaxf(sd[0], 1e-6f);
  if (act) kern[p*81+t] = kv/asum + 0.05f*pd[p*81+t];
}

// ---------------- K1: WMMA GEMM: fp1_w[8x768] x feature_map^T[768 x 16pix] ----
// A: M=channel (16 rows, 8 valid -> masked by multiply), K=768
// B: K x N=pixel(16). D: M=ch, N=pix. feature_map streamed with NT loads.
__global__ __launch_bounds__(256) void k_gemm1(
    const float* __restrict__ fm, const float* __restrict__ w1,
    const float* __restrict__ b1, float* __restrict__ t1){
  int lane = threadIdx.x & 31;
  int wv   = threadIdx.x >> 5;
  int pixbase = blockIdx.x*128 + wv*16;
  int bb = pixbase / HWn;
  int sp = pixbase % HWn;
  int ml = lane & 15;
  int hi = lane >> 4;
  float zm = (ml < RCn) ? 1.0f : 0.0f;                   // zero-pad rows 8..15
  const float* wrow = w1 + (size_t)(ml & 7)*FCn;         // always-valid address
  const float* fmb  = fm + (size_t)bb*FCn*HWn + sp + ml;
  v8f c = {0.f,0.f,0.f,0.f,0.f,0.f,0.f,0.f};
  for (int k0 = 0; k0 < FCn; k0 += 4){
    int ka = k0 + 2*hi;
    float2 wvv = *reinterpret_cast<const float2*>(wrow + ka);   // b64, aligned
    v2f a, b;
    a.x = wvv.x * zm;
    a.y = wvv.y * zm;
    b.x = __builtin_nontemporal_load(fmb + (size_t)ka*HWn);
    b.y = __builtin_nontemporal_load(fmb + (size_t)(ka+1)*HWn);
    c = __builtin_amdgcn_wmma_f32_16x16x4_f32(false, a, false, b, (short)0, c,
                                              false, false);
  }
  if (hi == 0){  // lanes 0-15 hold M=0..7 (valid channels), pixel N=ml
    float* o = t1 + (size_t)(pixbase+ml)*RCn;
    #pragma unroll
    for (int r = 0; r < 8; ++r) o[r] = geluf(c[r] + b1[r]);
  }
}

// ---------------- K2: 3x3 conv RC->RC + gelu; channel mean and |.| mean -----
__global__ void k_conv3(const float* __restrict__ t1, const float* __restrict__ w2,
    const float* __restrict__ b2, float* __restrict__ carrier,
    float* __restrict__ cm, float* __restrict__ am){
  int g = blockIdx.x*blockDim.x + threadIdx.x;
  if (g >= NPIX) return;
  int bb = g / HWn, sp = g % HWn, y = sp / Wn, x = sp % Wn;
  float acc[8];
  #pragma unroll
  for (int o=0;o<8;++o) acc[o] = b2[o];
  for (int oy=-1; oy<=1; ++oy){
    int yy = y+oy; if (yy<0||yy>=Hn) continue;
    for (int ox=-1; ox<=1; ++ox){
      int xx = x+ox; if (xx<0||xx>=Wn) continue;
      const float* tp = t1 + (size_t)(bb*HWn + yy*Wn + xx)*RCn;
      int kidx = (oy+1)*3 + (ox+1);
      #pragma unroll
      for (int ci=0; ci<8; ++ci){
        float v = tp[ci];
        #pragma unroll
        for (int o=0;o<8;++o) acc[o] += v * w2[(o*RCn+ci)*9 + kidx];
      }
    }
  }
  float s=0.f, sa=0.f;
  float* cp = carrier + (size_t)g*RCn;
  #pragma unroll
  for (int o=0;o<8;++o){ float v = geluf(acc[o]); cp[o]=v; s+=v; sa+=fabsf(v); }
  cm[g] = s*0.125f;
  am[g] = sa*0.125f;
}

// ---------------- avg3 helper kernel: mode 0 -> in - avg3(in); mode 1 -> avg3(|in|)
__global__ void k_avg3(const float* __restrict__ in, float* __restrict__ out, int mode){
  int g = blockIdx.x*blockDim.x + threadIdx.x;
  if (g >= NPIX) return;
  int bb = g / HWn, sp = g % HWn, y = sp / Wn, x = sp % Wn;
  float s = 0.f;
  for (int oy=-1; oy<=1; ++oy){
    int yy = y+oy; if (yy<0||yy>=Hn) continue;
    for (int ox=-1; ox<=1; ++ox){
      int xx = x+ox; if (xx<0||xx>=Wn) continue;
      float v = in[bb*HWn + yy*Wn + xx];
      s += (mode==1) ? fabsf(v) : v;
    }
  }
  s *= (1.0f/9.0f);
  out[g] = (mode==0) ? (in[g] - s) : s;
}

// ---------------- per-image sum/min/max reduction over HW ----------------
__global__ void k_reduce(const float* __restrict__ in, float* __restrict__ sum,
                         float* __restrict__ mn, float* __restrict__ mx){
  __shared__ float ss[256], sn[256], sx[256];
  int bb = blockIdx.x, t = threadIdx.x;
  float s=0.f, a=3.4e38f, b=-3.4e38f;
  for (int i=t; i<HWn; i+=256){
    float v = in[bb*HWn+i]; s+=v; a=fminf(a,v); b=fmaxf(b,v);
  }
  ss[t]=s; sn[t]=a; sx[t]=b; __syncthreads();
  for (int k=128; k>0; k>>=1){
    if (t<k){ ss[t]+=ss[t+k]; sn[t]=fminf(sn[t],sn[t+k]); sx[t]=fmaxf(sx[t],sx[t+k]); }
    __syncthreads();
  }
  if (t==0){ sum[bb]=ss[0]; mn[bb]=sn[0]; mx[bb]=sx[0]; }
}

// ---------------- K7: softmax+top2 prototypes, 9x9 responses, ev/gate seeds --
__global__ void k_proto(const float* __restrict__ theta, const float* __restrict__ length,
    const float* __restrict__ width, const float* __restrict__ plog,
    const float* __restrict__ chp, const float* __restrict__ kern,
    const float* __restrict__ am, const float* __restrict__ csraw,
    const float* __restrict__ aSum, const float* __restrict__ aMin,
    const float* __restrict__ aMax, const float* __restrict__ cMin,
    const float* __restrict__ cMax,
    float* __restrict__ ssraw, float* __restrict__ g0, float* __restrict__ m0){
  int g = blockIdx.x*blockDim.x + threadIdx.x;
  if (g >= NPIX) return;
  int bb = g / HWn, sp = g % HWn, y = sp / Wn, x = sp % Wn;
  float ta = tanhf(theta[g])*PI_F;
  float lv = sigmf_(length[g])*0.85f + 0.25f;
  float wv = sigmf_(width[g])*0.22f + 0.08f;
  const float Ls[3]={0.45f,0.75f,1.05f};
  const float Ws[3]={0.14f,0.20f,0.28f};
  float bias[8]; float bmax = -3.4e38f;
  #pragma unroll
  for (int p=0;p<8;++p){
    float ori = 2.0f*PI_F*(float)p/8.0f;
    float la = Ls[p%3], wa = Ws[(p/3)%3];
    float v = plog[((size_t)bb*Pn+p)*HWn + sp]
            + 1.25f*cosf(ta-ori)
            - (lv-la)*(lv-la)*(1.0f/0.08f)
            - (wv-wa)*(wv-wa)*(1.0f/0.01f);
    bias[p]=v; bmax=fmaxf(bmax,v);
  }
  float e[8], s=0.f;
  #pragma unroll
  for (int p=0;p<8;++p){ e[p]=expf(bias[p]-bmax); s+=e[p]; }
  int i1=0; float v1=-1.f;
  #pragma unroll
  for (int p=0;p<8;++p) if (e[p] > v1){ v1=e[p]; i1=p; }
  int i2=0; float v2=-1.f;
  #pragma unroll
  for (int p=0;p<8;++p) if (p!=i1 && e[p] > v2){ v2=e[p]; i2=p; }
  float pw1 = v1/s, pw2 = v2/s;
  float nrm = fmaxf(pw1+pw2, 1e-6f);
  pw1 /= nrm; pw2 /= nrm;
  // 9x9 correlations for the two selected prototypes only
  float r1=0.f, r2=0.f;
  const float* k1p = kern + i1*81;
  const float* k2p = kern + i2*81;
  for (int ky=0; ky<9; ++ky){
    int yy = y+ky-4; if (yy<0||yy>=Hn) continue;
    const float* row = chp + bb*HWn + yy*Wn;
    for (int kx=0; kx<9; ++kx){
      int xx = x+kx-4; if (xx<0||xx>=Wn) continue;
      float v = row[xx];
      r1 += v * k1p[ky*9+kx];
      r2 += v * k2p[ky*9+kx];
    }
  }
  ssraw[g] = r1*pw1 + r2*pw2;
  // support evidence
  float denom = fmaxf(aSum[bb]*(1.0f/(float)HWn), 1e-6f);
  float dens = am[g]/denom;
  float dmn = aMin[bb]/denom, dmx = aMax[bb]/denom;
  float dn = (dens - dmn)/fmaxf(dmx-dmn, 1e-6f);
  float cn = (csraw[g]-cMin[bb])/fmaxf(cMax[bb]-cMin[bb], 1e-6f);
  float ev = 0.65f*cn + 0.35f*dn;
  g0[g] = sigmf_((ev-0.2f)*(1.0f/0.08f));
  m0[g] = (ev >= 0.2f) ? 1.0f : 0.0f;
}

// ---------------- 5x5 max/min pool (OOB ignored, matches -inf/+inf init) ----
__global__ void k_pool5(const float* __restrict__ in, float* __restrict__ out, int isMin){
  int g = blockIdx.x*blockDim.x + threadIdx.x;
  if (g >= NPIX) return;
  int bb = g / HWn, sp = g % HWn, y = sp / Wn, x = sp % Wn;
  float r = isMin ? 3.4e38f : -3.4e38f;
  for (int oy=-2; oy<=2; ++oy){
    int yy = y+oy; if (yy<0||yy>=Hn) continue;
    for (int ox=-2; ox<=2; ++ox){
      int xx = x+ox; if (xx<0||xx>=Wn) continue;
      float v = in[bb*HWn + yy*Wn + xx];
      r = isMin ? fminf(r,v) : fmaxf(r,v);
    }
  }
  out[g] = r;
}

// ---------------- K12: final mask dilate, gate, blob --------------------------
__global__ void k_gateblob(const float* __restrict__ obuf, const float* __restrict__ gateb,
    const float* __restrict__ objn, float* __restrict__ blobg){
  int g = blockIdx.x*blockDim.x + threadIdx.x;
  if (g >= NPIX) return;
  int bb = g / HWn, sp = g % HWn, y = sp / Wn, x = sp % Wn;
  float mf = -3.4e38f;
  for (int oy=-2; oy<=2; ++oy){
    int yy = y+oy; if (yy<0||yy>=Hn) continue;
    for (int ox=-2; ox<=2; ++ox){
      int xx = x+ox; if (xx<0||xx>=Wn) continue;
      mf = fmaxf(mf, obuf[bb*HWn + yy*Wn + xx]);
    }
  }
  mf = fminf(fmaxf(mf, 0.0f), 1.0f);
  float s = 0.f;
  for (int oy=-1; oy<=1; ++oy){
    int yy = y+oy; if (yy<0||yy>=Hn) continue;
    for (int ox=-1; ox<=1; ++ox){
      int xx = x+ox; if (xx<0||xx>=Wn) continue;
      s += sigmf_(objn[bb*HWn + yy*Wn + xx]);
    }
  }
  blobg[g] = (s*(1.0f/9.0f)) * gateb[g] * mf;
}

// ---------------- K13: stroke gate, feats(12), pm1 (12->8) + gelu ------------
__global__ void k_feats(const float* __restrict__ ssb, const float* __restrict__ blobg,
    const float* __restrict__ am, const float* __restrict__ aSum,
    const float* __restrict__ alpha, const float* __restrict__ curv,
    const float* __restrict__ theta, const float* __restrict__ carrier,
    const float* __restrict__ wp1, const float* __restrict__ bp1,
    float* __restrict__ hbuf){
  int g = blockIdx.x*blockDim.x + threadIdx.x;
  if (g >= NPIX) return;
  int bb = g / HWn;
  float denom = fmaxf(aSum[bb]*(1.0f/(float)HWn), 1e-6f);
  float dens = am[g]/denom;
  float ag = sigmf_(alpha[g]);
  float cv = tanhf(curv[g]);
  float sg = ag * blobg[g] * (0.7f + 0.3f*fminf(fmaxf(dens,0.f),2.f));
  float ssf = sg * ssb[g] * (1.0f + 0.15f*cv);
  float ta = tanhf(theta[g])*PI_F;
  float dx = cosf(ta), dy = sinf(ta);
  float feats[12];
  const float* cp = carrier + (size_t)g*RCn;
  #pragma unroll
  for (int c=0;c<8;++c) feats[c] = cp[c]*ssf;
  feats[8]=ssf; feats[9]=dx*ssf; feats[10]=dy*ssf; feats[11]=cv*ssf;
  float* hp = hbuf + (size_t)g*RCn;
  #pragma unroll
  for (int o=0;o<8;++o){
    float a = bp1[o];
    #pragma unroll
    for (int c=0;c<12;++c) a += wp1[o*12+c]*feats[c];
    hp[o] = geluf(a);
  }
}

// ---------------- K14: WMMA GEMM: pm2_w[768x8] x h^T[8 x 16pix] + bias -------
// Output (100 MB) streamed with non-temporal stores, coalesced 64B rows.
__global__ __launch_bounds__(256) void k_gemm2(const float* __restrict__ hbuf,
    const float* __restrict__ w3, const float* __restrict__ b3,
    float* __restrict__ out){
  int lane = threadIdx.x & 31;
  int wv   = threadIdx.x >> 5;
  int pixbase = blockIdx.x*128 + wv*16;
  int bb = pixbase / HWn;
  int sp = pixbase % HWn;
  int ml = lane & 15;
  int hi = lane >> 4;
  const float* hp = hbuf + (size_t)(pixbase+ml)*RCn + 2*hi;
  float2 h0 = *reinterpret_cast<const float2*>(hp);       // K = 0..3 slice
  float2 h1 = *reinterpret_cast<const float2*>(hp + 4);   // K = 4..7 slice
  v2f bf0, bf1;
  bf0.x = h0.x; bf0.y = h0.y;
  bf1.x = h1.x; bf1.y = h1.y;
  float* ob = out + (size_t)bb*FCn*HWn + sp + ml;
  for (int ct=0; ct<FCn/16; ++ct){
    int chb = ct*16;
    const float* wrow = w3 + (size_t)(chb+ml)*RCn + 2*hi;
    float2 wa0 = *reinterpret_cast<const float2*>(wrow);
    float2 wa1 = *reinterpret_cast<const float2*>(wrow + 4);
    v8f c = {0.f,0.f,0.f,0.f,0.f,0.f,0.f,0.f};
    v2f a;
    a.x = wa0.x; a.y = wa0.y;
    c = __builtin_amdgcn_wmma_f32_16x16x4_f32(false, a, false, bf0, (short)0, c,
                                              false, false);
    a.x = wa1.x; a.y = wa1.y;
    c = __builtin_amdgcn_wmma_f32_16x16x4_f32(false, a, false, bf1, (short)0, c,
                                              false, false);
    // D: lane holds pixel N=ml, channels M = 8*hi + r  -> coalesced row stores
    #pragma unroll
    for (int r=0;r<8;++r){
      int ch = chb + 8*hi + r;
      __builtin_nontemporal_store(c[r] + b3[ch], ob + (size_t)ch*HWn);
    }
  }
}

extern "C" void kernel_launch(void* const* d_in, const int* in_sizes, int n_in,
                              void* d_out, int out_size, void* d_ws, size_t ws_size,
                              hipStream_t stream) {
  const float* fm    = (const float*)d_in[0];
  const float* theta = (const float*)d_in[1];
  const float* len   = (const float*)d_in[2];
  const float* wid   = (const float*)d_in[3];
  const float* curv  = (const float*)d_in[4];
  const float* alpha = (const float*)d_in[5];
  const float* objn  = (const float*)d_in[6];
  const float* plog  = (const float*)d_in[7];
  const float* w1    = (const float*)d_in[8];
  const float* b1    = (const float*)d_in[9];
  const float* w2    = (const float*)d_in[10];
  const float* b2    = (const float*)d_in[11];
  const float* wp1   = (const float*)d_in[12];
  const float* bp1   = (const float*)d_in[13];
  const float* w3    = (const float*)d_in[14];
  const float* b3    = (const float*)d_in[15];
  const float* pd    = (const float*)d_in[16];

  float* ws      = (float*)d_ws;
  float* t1      = ws;                          // NPIX*8
  float* carrier = t1      + (size_t)NPIX*RCn;  // NPIX*8
  float* hbuf    = carrier + (size_t)NPIX*RCn;  // NPIX*8
  float* cmb     = hbuf    + (size_t)NPIX*RCn;
  float* chp     = cmb     + NPIX;
  float* amb     = chp     + NPIX;
  float* csr     = amb     + NPIX;
  float* ssraw   = csr     + NPIX;
  float* ssb     = ssraw   + NPIX;
  float* g0      = ssb     + NPIX;
  float* m0b     = g0      + NPIX;
  float* gateb   = m0b     + NPIX;
  float* ebuf    = gateb   + NPIX;
  float* obuf    = ebuf    + NPIX;
  float* blobg   = obuf    + NPIX;
  float* kern    = blobg   + NPIX;              // 648
  float* red     = kern    + 648;               // 48
  float* aSum = red;      float* aMin = red+8;  float* aMax = red+16;
  float* cMin = red+24;   float* cMax = red+32; float* cDum = red+40;

  const int TB = 256;
  const int pixBlocks = NPIX/TB;   // 128

  k_bank <<<Pn, 128, 0, stream>>>(pd, kern);
  k_gemm1<<<NPIX/128, TB, 0, stream>>>(fm, w1, b1, t1);
  k_conv3<<<pixBlocks, TB, 0, stream>>>(t1, w2, b2, carrier, cmb, amb);
  k_avg3 <<<pixBlocks, TB, 0, stream>>>(cmb, chp, 0);      // carrier_hp
  k_avg3 <<<pixBlocks, TB, 0, stream>>>(chp, csr, 1);      // avg3(|chp|)
  k_reduce<<<Bn, TB, 0, stream>>>(amb, aSum, aMin, aMax);
  k_reduce<<<Bn, TB, 0, stream>>>(csr, cDum, cMin, cMax);
  k_proto<<<pixBlocks, TB, 0, stream>>>(theta, len, wid, plog, chp, kern,
        amb, csr, aSum, aMin, aMax, cMin, cMax, ssraw, g0, m0b);
  k_avg3 <<<pixBlocks, TB, 0, stream>>>(ssraw, ssb, 0);    // ss highpass
  k_pool5<<<pixBlocks, TB, 0, stream>>>(g0,  gateb, 0);    // dilate gate
  k_pool5<<<pixBlocks, TB, 0, stream>>>(m0b, ebuf,  1);    // erode mask
  k_pool5<<<pixBlocks, TB, 0, stream>>>(ebuf, obuf, 0);    // open
  k_gateblob<<<pixBlocks, TB, 0, stream>>>(obuf, gateb, objn, blobg);
  k_feats<<<pixBlocks, TB, 0, stream>>>(ssb, blobg, amb, aSum, alpha, curv,
        theta, carrier, wp1, bp1, hbuf);
  k_gemm2<<<NPIX/128, TB, 0, stream>>>(hbuf, w3, b3, (float*)d_out);
}